// ResGatedConv_46712064311852
// MI455X (gfx1250) — compile-verified
//
#include <hip/hip_runtime.h>
#include <math.h>

#define NNODES 100000
#define NEDGES 1600000
#define D      128
#define NF     (NNODES * D)          // 12,800,000 floats per feature matrix

typedef float v2f __attribute__((ext_vector_type(2)));
typedef float v8f __attribute__((ext_vector_type(8)));

// ---------------------------------------------------------------------------
// Kernel 1: zero the degree + fill counters (2 * NNODES ints)
// ---------------------------------------------------------------------------
__global__ __launch_bounds__(256) void zero_int_kernel(int* __restrict__ p, int n) {
    int i = blockIdx.x * blockDim.x + threadIdx.x;
    if (i < n) p[i] = 0;
}

// ---------------------------------------------------------------------------
// Kernel 2: fused 4-matrix GEMM with V_WMMA_F32_16X16X4_F32.
//   x1  = x@W1.T + b1                   -> d_out
//   Y2  = 2*(x@W2.T + b2)               -> ws
//   Y34 = 2*(x@W3.T+b3) + 2*(x@W4.T+b4) -> ws   (same edge list => fuse)
// One block = 16 nodes; 8 waves each own a 16-wide output-feature tile.
// ---------------------------------------------------------------------------
__global__ __launch_bounds__(256)
void gemm4_wmma_kernel(const float* __restrict__ x,
                       const float* __restrict__ W1, const float* __restrict__ b1,
                       const float* __restrict__ W2, const float* __restrict__ b2,
                       const float* __restrict__ W3, const float* __restrict__ b3,
                       const float* __restrict__ W4, const float* __restrict__ b4,
                       float* __restrict__ x1_out,
                       float* __restrict__ Y2,
                       float* __restrict__ Y34)
{
    __shared__ float xs[16][132];                 // pad 132 -> conflict-free column reads
    const int tid   = threadIdx.x;
    const int node0 = blockIdx.x * 16;            // 100000 % 16 == 0, no tail

    #pragma unroll
    for (int i = 0; i < 8; ++i) {                 // stage 16x128 x-tile, coalesced
        int idx = tid + i * 256;
        int r = idx >> 7;
        int c = idx & 127;
        xs[r][c] = x[(node0 + r) * D + c];
    }
    __syncthreads();

    const int wave = tid >> 5;                    // 0..7 -> output-feature tile
    const int lane = tid & 31;
    const int m    = lane & 15;                   // A row / B,C column within tile
    const int kh   = (lane >> 4) << 1;            // K sub-offset: 0 or 2
    const int n    = wave * 16 + m;               // global output feature column

    const float bb1 = b1[n], bb2 = b2[n], bb3 = b3[n], bb4 = b4[n];
    v8f acc1 = {bb1,bb1,bb1,bb1,bb1,bb1,bb1,bb1};
    v8f acc2 = {bb2,bb2,bb2,bb2,bb2,bb2,bb2,bb2};
    v8f acc3 = {bb3,bb3,bb3,bb3,bb3,bb3,bb3,bb3};
    v8f acc4 = {bb4,bb4,bb4,bb4,bb4,bb4,bb4,bb4};

    const float* w1p = W1 + n * D;
    const float* w2p = W2 + n * D;
    const float* w3p = W3 + n * D;
    const float* w4p = W4 + n * D;

    #pragma unroll 4
    for (int k0 = 0; k0 < D; k0 += 4) {
        const v2f a   = *(const v2f*)&xs[m][k0 + kh];       // A[m][k0+kh+j]
        const v2f bw1 = *(const v2f*)&w1p[k0 + kh];         // B[k][n] = W[n][k]
        const v2f bw2 = *(const v2f*)&w2p[k0 + kh];
        const v2f bw3 = *(const v2f*)&w3p[k0 + kh];
        const v2f bw4 = *(const v2f*)&w4p[k0 + kh];
        acc1 = __builtin_amdgcn_wmma_f32_16x16x4_f32(false, a, false, bw1, (short)0, acc1, false, false);
        acc2 = __builtin_amdgcn_wmma_f32_16x16x4_f32(false, a, false, bw2, (short)0, acc2, false, false);
        acc3 = __builtin_amdgcn_wmma_f32_16x16x4_f32(false, a, false, bw3, (short)0, acc3, false, false);
        acc4 = __builtin_amdgcn_wmma_f32_16x16x4_f32(false, a, false, bw4, (short)0, acc4, false, false);
    }

    // C/D layout: vgpr j -> row (j + 8*(lane>>4)), col (lane&15)
    const int rbase = node0 + ((lane >> 4) << 3);
    #pragma unroll
    for (int j = 0; j < 8; ++j) {
        const int off = (rbase + j) * D + n;
        x1_out[off] = acc1[j];
        Y2[off]     = 2.0f * acc2[j];                  // fold the (y+y) doubling
        Y34[off]    = 2.0f * (acc3[j] + acc4[j]);      // fused W3+W4 messages
    }
}

// ---------------------------------------------------------------------------
// Kernel 3: destination-degree histogram
// ---------------------------------------------------------------------------
__global__ __launch_bounds__(256)
void hist_kernel(const int* __restrict__ dst, int* __restrict__ deg) {
    int e = blockIdx.x * blockDim.x + threadIdx.x;
    if (e < NEDGES) atomicAdd(&deg[dst[e]], 1);
}

// ---------------------------------------------------------------------------
// Kernel 4: single-block exclusive scan of deg -> rowptr (100001 entries).
// Chunked Hillis-Steele, 4 elements per thread per chunk (25 chunk iters).
// ---------------------------------------------------------------------------
#define SCAN_T 1024
__global__ __launch_bounds__(SCAN_T)
void scan_kernel(const int* __restrict__ deg, int* __restrict__ rowptr) {
    __shared__ int buf[SCAN_T];
    __shared__ int carry_s;
    const int tid = threadIdx.x;
    if (tid == 0) carry_s = 0;
    __syncthreads();
    for (int base = 0; base < NNODES; base += SCAN_T * 4) {
        const int i0 = base + tid * 4;
        const int v0 = (i0 + 0 < NNODES) ? deg[i0 + 0] : 0;
        const int v1 = (i0 + 1 < NNODES) ? deg[i0 + 1] : 0;
        const int v2 = (i0 + 2 < NNODES) ? deg[i0 + 2] : 0;
        const int v3 = (i0 + 3 < NNODES) ? deg[i0 + 3] : 0;
        const int s  = v0 + v1 + v2 + v3;
        buf[tid] = s;
        __syncthreads();
        for (int ofs = 1; ofs < SCAN_T; ofs <<= 1) {
            int t = (tid >= ofs) ? buf[tid - ofs] : 0;
            __syncthreads();
            buf[tid] += t;
            __syncthreads();
        }
        int excl = carry_s + buf[tid] - s;         // exclusive prefix of elem 0
        if (i0 + 0 < NNODES) rowptr[i0 + 0] = excl;  excl += v0;
        if (i0 + 1 < NNODES) rowptr[i0 + 1] = excl;  excl += v1;
        if (i0 + 2 < NNODES) rowptr[i0 + 2] = excl;  excl += v2;
        if (i0 + 3 < NNODES) rowptr[i0 + 3] = excl;
        __syncthreads();
        if (tid == SCAN_T - 1) carry_s += buf[tid];
        __syncthreads();
    }
    if (tid == 0) rowptr[NNODES] = carry_s;        // == NEDGES
}

// ---------------------------------------------------------------------------
// Kernel 5: bucket-fill edge sources by destination (CSR column array)
// ---------------------------------------------------------------------------
__global__ __launch_bounds__(256)
void fill_kernel(const int* __restrict__ src, const int* __restrict__ dst,
                 const int* __restrict__ rowptr, int* __restrict__ fill,
                 int* __restrict__ esrc) {
    int e = blockIdx.x * blockDim.x + threadIdx.x;
    if (e >= NEDGES) return;
    const int d = dst[e];
    const int pos = rowptr[d] + atomicAdd(&fill[d], 1);
    esrc[pos] = src[e];
}

// ---------------------------------------------------------------------------
// Kernel 6: per-destination gather + fused gate:
//   out[n] = sigmoid(sum Y34[src]) * (sum Y2[src]) + x1[n]
// One wave per node; lane owns a float4 column slice. esrc[e] is
// wave-uniform -> scalar load; Y2/Y34 (102 MB) are L2-resident on MI455X.
// ---------------------------------------------------------------------------
__global__ __launch_bounds__(256)
void gather_gate_kernel(const int* __restrict__ rowptr, const int* __restrict__ esrc,
                        const float* __restrict__ Y2, const float* __restrict__ Y34,
                        float* __restrict__ out)
{
    const int wave = threadIdx.x >> 5;
    const int lane = threadIdx.x & 31;
    const int n = blockIdx.x * 8 + wave;          // 12500*8 == 100000, no tail
    const int e0 = rowptr[n];
    const int e1 = rowptr[n + 1];
    const int col = lane * 4;

    float4 a2  = make_float4(0.f, 0.f, 0.f, 0.f);
    float4 a34 = make_float4(0.f, 0.f, 0.f, 0.f);
    for (int e = e0; e < e1; ++e) {
        const int s = esrc[e];
        const float4 y2  = *(const float4*)(Y2  + s * D + col);
        const float4 y34 = *(const float4*)(Y34 + s * D + col);
        a2.x  += y2.x;   a2.y  += y2.y;   a2.z  += y2.z;   a2.w  += y2.w;
        a34.x += y34.x;  a34.y += y34.y;  a34.z += y34.z;  a34.w += y34.w;
    }

    const int o = n * D + col;
    const float4 x1 = *(const float4*)(out + o);  // written by GEMM kernel
    float4 r;
    r.x = a2.x / (1.0f + __expf(-a34.x)) + x1.x;
    r.y = a2.y / (1.0f + __expf(-a34.y)) + x1.y;
    r.z = a2.z / (1.0f + __expf(-a34.z)) + x1.z;
    r.w = a2.w / (1.0f + __expf(-a34.w)) + x1.w;
    *(float4*)(out + o) = r;
}

// ---------------------------------------------------------------------------
extern "C" void kernel_launch(void* const* d_in, const int* in_sizes, int n_in,
                              void* d_out, int out_size, void* d_ws, size_t ws_size,
                              hipStream_t stream) {
    const float* x        = (const float*)d_in[0];
    const int*   edge_idx = (const int*)  d_in[1];
    const float* W1 = (const float*)d_in[2];  const float* b1 = (const float*)d_in[3];
    const float* W2 = (const float*)d_in[4];  const float* b2 = (const float*)d_in[5];
    const float* W3 = (const float*)d_in[6];  const float* b3 = (const float*)d_in[7];
    const float* W4 = (const float*)d_in[8];  const float* b4 = (const float*)d_in[9];

    const int* src = edge_idx;               // edge_idx[0, :]
    const int* dst = edge_idx + NEDGES;      // edge_idx[1, :]

    float* wsf = (float*)d_ws;
    float* Y2  = wsf;                        // NF floats
    float* Y34 = wsf + (size_t)NF;           // NF floats
    int*   wsi    = (int*)(wsf + 2 * (size_t)NF);
    int*   deg    = wsi;                     // NNODES
    int*   fill   = wsi + NNODES;            // NNODES
    int*   rowptr = wsi + 2 * NNODES;        // NNODES + 1
    int*   esrc   = wsi + 3 * NNODES + 1;    // NEDGES

    float* out = (float*)d_out;

    // 1) zero deg + fill (contiguous 2*NNODES ints)
    zero_int_kernel<<<(2 * NNODES + 255) / 256, 256, 0, stream>>>(deg, 2 * NNODES);
    // 2) fused quad-GEMM via WMMA: x1 -> d_out, Y2/Y34 -> ws
    gemm4_wmma_kernel<<<NNODES / 16, 256, 0, stream>>>(
        x, W1, b1, W2, b2, W3, b3, W4, b4, out, Y2, Y34);
    // 3) destination-degree histogram
    hist_kernel<<<(NEDGES + 255) / 256, 256, 0, stream>>>(dst, deg);
    // 4) exclusive scan -> rowptr
    scan_kernel<<<1, SCAN_T, 0, stream>>>(deg, rowptr);
    // 5) CSR bucket fill
    fill_kernel<<<(NEDGES + 255) / 256, 256, 0, stream>>>(src, dst, rowptr, fill, esrc);
    // 6) gather + sigmoid gate + residual (fused epilogue)
    gather_gate_kernel<<<NNODES / 8, 256, 0, stream>>>(rowptr, esrc, Y2, Y34, out);
}